// Smoother_6906307412277
// MI455X (gfx1250) — compile-verified
//
#include <hip/hip_runtime.h>
#include <cstdint>
#include <cstddef>

// Problem constants (from the reference)
#define K_LEN  501
#define PAD    250
#define B_DIM  8
#define T_DIM  16384
#define C_DIM  128

// Tiling
#define CG       16                    // channels per block
#define L_OUT    256                   // output timesteps per block
#define ROWS     (L_OUT + K_LEN - 1)   // 756 staged input rows (tile + halo)
#define SEGS     16                    // scan segments per channel
#define SEGLEN   49                    // ceil(756/16)=48, padded to 49 (bank spread)
#define ROWS_AL  (SEGS * SEGLEN)       // 784 allocated rows
#define RS       20                    // LDS row stride in floats (16 data + 4 pad; keeps
                                       // 16B async chunks aligned, avoids bank conflicts)

typedef int v4i __attribute__((vector_size(16)));              // v4i32, 16 bytes
typedef __attribute__((address_space(1))) v4i* gptr_v4i;       // global
typedef __attribute__((address_space(3))) v4i* lptr_v4i;       // LDS

// ---- CDNA5 async global->LDS copy (16 bytes per lane) ----
__device__ __forceinline__ void async_g2l_b128(const float* gsrc, float* ldst) {
#if __has_builtin(__builtin_amdgcn_global_load_async_to_lds_b128)
    __builtin_amdgcn_global_load_async_to_lds_b128(
        (gptr_v4i)(uintptr_t)gsrc,
        (lptr_v4i)(uintptr_t)ldst,
        0, 0);
#else
    uint32_t lds_off = (uint32_t)(uintptr_t)ldst;   // low 32 bits = LDS byte offset
    uint64_t gaddr   = (uint64_t)(uintptr_t)gsrc;
    asm volatile("global_load_async_to_lds_b128 %0, %1, off"
                 :: "v"(lds_off), "v"(gaddr) : "memory");
#endif
}

__device__ __forceinline__ void wait_async0() {
#if __has_builtin(__builtin_amdgcn_s_wait_asynccnt)
    __builtin_amdgcn_s_wait_asynccnt(0);
#else
    asm volatile("s_wait_asynccnt 0" ::: "memory");
#endif
}

__device__ __forceinline__ int mirror_t(int t) {
    t = (t < 0) ? -t : t;                     // reflect (no edge repeat), P < T so one bounce
    t = (t >= T_DIM) ? (2 * (T_DIM - 1) - t) : t;
    return t;
}

__global__ __launch_bounds__(256) void
smoother_box501_kernel(const float* __restrict__ pred,
                       const float* __restrict__ bias,
                       float* __restrict__ out)
{
    __shared__ __align__(16) float tile[ROWS_AL * RS];   // 784*20*4 = 62,720 B
    __shared__ float segsum[CG * SEGS];                  //           1,024 B

    const int chunk = blockIdx.x;          // 0..63  (T/L_OUT)
    const int cgrp  = blockIdx.y;          // 0..7   (C/CG)
    const int b     = blockIdx.z;          // 0..7
    const int t0    = chunk * L_OUT;
    const int cbase = cgrp * CG;
    const int tid   = (int)threadIdx.x;

    // ---------------- Phase 1: async stage rows [t0-250, t0+505] into LDS ---------
    {
        const int q   = tid & 3;           // which 16B sub-chunk of the 64B row
        const int rw0 = tid >> 2;          // 64 rows covered per iteration
        for (int it = 0; it < (ROWS_AL / 64); ++it) {   // 12 iterations (768 >= 756)
            const int row = rw0 + it * 64;
            if (row < ROWS) {
                const int t = mirror_t(t0 - PAD + row);
                const float* g = pred + ((size_t)(b * T_DIM + t) * C_DIM + cbase + q * 4);
                float* l = &tile[row * RS + q * 4];
                async_g2l_b128(g, l);
            }
        }
    }
    wait_async0();          // this wave's async copies are in LDS
    __syncthreads();        // all waves' copies visible

    // ---------------- Phase 2: per-channel inclusive prefix sum over 756 rows -----
    const int c   = tid & (CG - 1);        // channel within tile
    const int seg = tid >> 4;              // scan segment
    const int r0  = seg * SEGLEN;

    float ssum = 0.0f;
    #pragma unroll 7
    for (int i = 0; i < SEGLEN; ++i) {
        const int r = r0 + i;
        if (r < ROWS) ssum += tile[r * RS + c];
    }
    segsum[c * SEGS + seg] = ssum;
    __syncthreads();

    float run = 0.0f;
    for (int s = 0; s < seg; ++s) run += segsum[c * SEGS + s];   // exclusive offset
    #pragma unroll 7
    for (int i = 0; i < SEGLEN; ++i) {
        const int r = r0 + i;
        if (r < ROWS) {
            run += tile[r * RS + c];
            tile[r * RS + c] = run;        // inclusive prefix, disjoint per thread
        }
    }
    __syncthreads();

    // ---------------- Phase 3: windowed sums + max, streamed out ------------------
    const float bval  = bias[0];
    const float inv_k = 1.0f / (float)K_LEN;
    const int   og    = tid >> 4;          // 16 consecutive timesteps per thread

    #pragma unroll 4
    for (int i = 0; i < (L_OUT / SEGS); ++i) {
        const int   tloc = og * (L_OUT / SEGS) + i;
        const float hi   = tile[(tloc + (K_LEN - 1)) * RS + c];
        const float lo   = (tloc > 0) ? tile[(tloc - 1) * RS + c] : 0.0f;
        const float win  = hi - lo;
        const float sm   = (win + bval) * inv_k;

        const size_t gidx = (size_t)(b * T_DIM + (t0 + tloc)) * C_DIM + (cbase + c);
        const float  p    = pred[gidx];     // L2 hit; exact value for the max
        out[gidx] = fmaxf(p, sm);
    }
}

extern "C" void kernel_launch(void* const* d_in, const int* in_sizes, int n_in,
                              void* d_out, int out_size, void* d_ws, size_t ws_size,
                              hipStream_t stream) {
    (void)in_sizes; (void)n_in; (void)d_ws; (void)ws_size; (void)out_size;
    const float* pred = (const float*)d_in[0];   // [8, 16384, 128] f32
    const float* bias = (const float*)d_in[1];   // [1] f32
    float* out = (float*)d_out;                  // [8, 16384, 128] f32

    dim3 grid(T_DIM / L_OUT, C_DIM / CG, B_DIM); // 64 x 8 x 8 = 4096 blocks
    dim3 block(256);                             // 8 waves (wave32)
    smoother_box501_kernel<<<grid, block, 0, stream>>>(pred, bias, out);
}